// MoEMLP_19104014532974
// MI455X (gfx1250) — compile-verified
//
#include <hip/hip_runtime.h>
#include <hip/hip_bf16.h>
#include <math.h>
#include <stdint.h>

// ---------------------------------------------------------------------------
// MoE MLP (top-2 of 8 experts + shared expert), fp32, MI455X / gfx1250.
//
// Roofline: weights = 1.25 GB fp32 -> ~54us floor at 23.3 TB/s HBM.  Sparse
// FLOPs ~1.7e12 -> far below even the f32 WMMA ceiling.  => memory bound;
// keep exact fp32 semantics with V_WMMA_F32_16X16X4_F32 and fuse
// gate/up/silu/down so the (T*K+T) x I intermediate never touches HBM.
//
// CDNA5 data movement: the down-projection weight tile Wd[0:2048, ib:ib+16]
// (128 KB, needed by the whole workgroup) is DMA'd into LDS by the Tensor
// Data Mover (TENSOR_LOAD_TO_LDS, TENSORcnt) asynchronously, overlapped with
// the phase-1 WMMA K-loop.  TDM padding (16 DWORDs -> +2 DWORDs) gives an
// 18-float LDS row stride = bank-conflict-free ds_load_b64 B-reads.
// ---------------------------------------------------------------------------

#define E_   8
#define D_   2048
#define I_   5632
#define T_   8192
#define AUXC 0.01f

#define BM        16      // tokens per workgroup (one WMMA M tile)
#define THREADS   512     // 16 waves
#define IK        128     // intermediate-dim superchunk (8 sub-tiles of 16)
#define XS_STRIDE 2052    // 2048 padded; %64==4 -> conflict-free ds_load_b64
#define HS_STRIDE 132     // 128 padded; %64==4 -> conflict-free ds_load_b64
#define WD_STRIDE 18      // 16 + TDM pad(2 DWORDs / 16 DWORDs) -> conflict-free
#define WD_ROWS   2048

// LDS layout (floats)
#define OFF_GU   (BM * XS_STRIDE)
#define OFF_HS   (OFF_GU + 16 * 256)
#define OFF_WD   (OFF_HS + BM * HS_STRIDE)
#define OFF_CW   (OFF_WD + WD_ROWS * WD_STRIDE)
#define OFF_TOK  (OFF_CW + BM)
#define SMEM_FLOATS (OFF_TOK + BM)

typedef float v2f __attribute__((ext_vector_type(2)));
typedef float v8f __attribute__((ext_vector_type(8)));
typedef unsigned int u32x4 __attribute__((ext_vector_type(4)));
typedef int i32x4 __attribute__((ext_vector_type(4)));
typedef int i32x8 __attribute__((ext_vector_type(8)));

#if defined(__HIP_DEVICE_COMPILE__) && \
    __has_builtin(__builtin_amdgcn_tensor_load_to_lds) && \
    __has_builtin(__builtin_amdgcn_s_wait_tensorcnt)
#define USE_TDM 1
#else
#define USE_TDM 0
#endif

#if USE_TDM
typedef __attribute__((address_space(3))) void lds_void;
static __device__ inline uint32_t lds_offset_u32(const void* p) {
  return (uint32_t)(uintptr_t)(lds_void*)p;   // generic -> LDS offset
}
#endif

#define WMMA_F32X4(acc, a, b) \
  __builtin_amdgcn_wmma_f32_16x16x4_f32(false, (a), false, (b), (short)0, (acc), false, false)

// ---------------------------------------------------------------------------
// Zero output + routing counters (output is accumulated with f32 atomics).
// ---------------------------------------------------------------------------
__global__ void moe_zero_kernel(float* __restrict__ out, size_t n,
                                int* __restrict__ cnt, float* __restrict__ ps) {
  size_t i = (size_t)blockIdx.x * blockDim.x + threadIdx.x;
  size_t stride = (size_t)gridDim.x * blockDim.x;
  for (size_t k = i; k < n; k += stride) out[k] = 0.0f;
  if (blockIdx.x == 0 && threadIdx.x < E_) {
    cnt[threadIdx.x] = 0;
    ps[threadIdx.x] = 0.0f;
  }
}

// ---------------------------------------------------------------------------
// Router: one wave32 per token.  8 logits, top-2 + softmax(top-2), full
// softmax accumulation for the aux loss, per-expert gather lists.
// ---------------------------------------------------------------------------
__global__ __launch_bounds__(256)
void moe_router_kernel(const float* __restrict__ x, const float* __restrict__ gw,
                       int* __restrict__ cnt, float* __restrict__ ps,
                       int* __restrict__ idx, float* __restrict__ wl) {
  const int wave = threadIdx.x >> 5;
  const int lane = threadIdx.x & 31;
  const int t = blockIdx.x * 8 + wave;

  float acc[E_];
#pragma unroll
  for (int e = 0; e < E_; ++e) acc[e] = 0.0f;

  const float* xr = x + (size_t)t * D_;
  for (int j = 0; j < D_ / 32; ++j) {
    const int d = lane + j * 32;
    const float xv = xr[d];
#pragma unroll
    for (int e = 0; e < E_; ++e) acc[e] += xv * gw[e * D_ + d];
  }
#pragma unroll
  for (int e = 0; e < E_; ++e) {
#pragma unroll
    for (int off = 16; off > 0; off >>= 1) acc[e] += __shfl_xor(acc[e], off, 32);
  }

  if (lane == 0) {
    int e1 = 0; float v1 = acc[0];
    for (int e = 1; e < E_; ++e) if (acc[e] > v1) { v1 = acc[e]; e1 = e; }
    int e2 = (e1 == 0) ? 1 : 0; float v2 = acc[e2];
    for (int e = 0; e < E_; ++e) {
      if (e == e1) continue;
      if (acc[e] > v2) { v2 = acc[e]; e2 = e; }
    }
    const float r  = expf(v2 - v1);
    const float w1 = 1.0f / (1.0f + r);
    const float w2 = r * w1;

    float s = 0.0f, p[E_];
    for (int e = 0; e < E_; ++e) { p[e] = expf(acc[e] - v1); s += p[e]; }
    const float inv = 1.0f / s;
    for (int e = 0; e < E_; ++e) atomicAdd(&ps[e], p[e] * inv);

    int p1 = atomicAdd(&cnt[e1], 1);
    idx[e1 * T_ + p1] = t; wl[e1 * T_ + p1] = w1;
    int p2 = atomicAdd(&cnt[e2], 1);
    idx[e2 * T_ + p2] = t; wl[e2 * T_ + p2] = w2;
  }
}

__global__ void moe_aux_kernel(const int* __restrict__ cnt,
                               const float* __restrict__ ps,
                               float* __restrict__ aux_out) {
  if (threadIdx.x == 0 && blockIdx.x == 0) {
    float s = 0.0f;
    for (int e = 0; e < E_; ++e)
      s += ((float)cnt[e] / (float)T_) * (ps[e] / (float)T_);
    *aux_out = AUXC * s * (float)E_;
  }
}

// ---------------------------------------------------------------------------
// Fused expert MLP.  blockIdx.y = expert (0..7) or 8 = shared expert.
// blockIdx.x = 16-token tile.  512 threads = 16 waves.
// ---------------------------------------------------------------------------
__global__ __launch_bounds__(THREADS)
void moe_mlp_kernel(const float* __restrict__ x,
                    const float* __restrict__ eg, const float* __restrict__ eu,
                    const float* __restrict__ ed,
                    const float* __restrict__ sg, const float* __restrict__ su,
                    const float* __restrict__ sd,
                    const int* __restrict__ idx, const int* __restrict__ cnt,
                    const float* __restrict__ wl,
                    float* __restrict__ out) {
  extern __shared__ float smem[];
  float* Xs    = smem;                 // BM x XS_STRIDE
  float* GU    = smem + OFF_GU;        // 16 tiles x 256 (G even wave, U odd)
  float* Hs    = smem + OFF_HS;        // BM x HS_STRIDE
  float* WdLds = smem + OFF_WD;        // WD_ROWS x WD_STRIDE (TDM destination)
  float* cw_s  = smem + OFF_CW;        // BM combine weights
  int*   tok_s = (int*)(smem + OFF_TOK);

  const int e = blockIdx.y;
  const float *Wg, *Wu, *Wd;
  int nTok;
  if (e < E_) {
    nTok = cnt[e];
    if ((int)blockIdx.x * BM >= nTok) return;   // uniform early-out
    Wg = eg + (size_t)e * I_ * D_;
    Wu = eu + (size_t)e * I_ * D_;
    Wd = ed + (size_t)e * D_ * I_;
  } else {
    nTok = T_;
    Wg = sg; Wu = su; Wd = sd;
  }

  const int tid = threadIdx.x;
  if (tid < BM) {
    const int p = blockIdx.x * BM + tid;
    if (e < E_) {
      if (p < nTok) { tok_s[tid] = idx[e * T_ + p]; cw_s[tid] = wl[e * T_ + p]; }
      else          { tok_s[tid] = idx[e * T_];     cw_s[tid] = 0.0f; }  // pad row
    } else {
      tok_s[tid] = p; cw_s[tid] = 1.0f;
    }
  }
  __syncthreads();

  // Stage gathered X tile (16 x 2048) into LDS, float4, coalesced.
  {
    const int r = tid >> 5, l32 = tid & 31;
    const float4* xrow = (const float4*)(x + (size_t)tok_s[r] * D_);
    float4* xd = (float4*)(Xs + r * XS_STRIDE);
#pragma unroll
    for (int j = 0; j < D_ / (32 * 4); ++j) xd[l32 + j * 32] = xrow[l32 + j * 32];
  }
  __syncthreads();

  const int wave = tid >> 5;
  const int lane = tid & 31;
  const int mrow = lane & 15;            // M rows (A) or N cols (B/C)
  const int koff = (lane >> 4) << 1;     // 0 (lanes 0-15) / 2 (lanes 16-31)
  const int half = (lane >> 4) << 3;     // C-tile row offset 0/8

  v8f oacc[8];
#pragma unroll
  for (int i = 0; i < 8; ++i) oacc[i] = (v8f){0,0,0,0,0,0,0,0};

  const int sub = wave >> 1;                        // i-subchunk 0..7
  const float* W1 = (wave & 1) ? Wu : Wg;           // this wave's phase-1 matrix
  const float* xrowA = Xs + mrow * XS_STRIDE;

#if USE_TDM
  const uint32_t wdLdsAddr = lds_offset_u32(WdLds);
#endif

  for (int ib = 0; ib < I_; ib += IK) {
#if USE_TDM
    // ---- async TDM: DMA Wd[0:2048, ib:ib+16] -> LDS, overlapped w/ phase 1
    if (wave == 0) {
      const uint64_t ga = (uint64_t)(uintptr_t)(Wd + ib);   // tile start
      u32x4 g0;
      g0[0] = 1u;                                  // count=1 valid descriptor
      g0[1] = wdLdsAddr;                           // lds_addr
      g0[2] = (uint32_t)ga;                        // global_addr[31:0]
      g0[3] = (uint32_t)((ga >> 32) & 0x01FFFFFFu) // global_addr[56:32]
              | 0x80000000u;                       // type=2 ("image")
      i32x8 g1;
      g1[0] = (2 << 16)      // data_size = 4 bytes
            | (1 << 20)      // pad_enable
            | (3 << 22)      // pad_interval: 16 DWORDs
            | (1 << 25);     // pad_amount:   2 DWORDs  -> row stride 18 floats
      g1[1] = (int)((I_ & 0xFFFF) << 16);                       // tensor_dim0 lo
      g1[2] = (int)(((I_ >> 16) & 0xFFFF) | ((D_ & 0xFFFF) << 16)); // dim0 hi|dim1 lo
      g1[3] = (int)(((D_ >> 16) & 0xFFFF) | (16 << 16));        // dim1 hi|tile_dim0=16
      g1[4] = (int)(WD_ROWS & 0xFFFF);                          // tile_dim1=2048
      g1[5] = (int)I_;                                          // dim0_stride lo32
      g1[6] = 0;
      g1[7] = 0;
      i32x4 g2 = {0, 0, 0, 0}, g3 = {0, 0, 0, 0};
#if __clang_major__ >= 23
      i32x8 z8 = {0, 0, 0, 0, 0, 0, 0, 0};
      __builtin_amdgcn_tensor_load_to_lds(g0, g1, g2, g3, z8, 0);
#else
      __builtin_amdgcn_tensor_load_to_lds(g0, g1, g2, g3, 0);
#endif
    }
#endif

    // ---------- phase 1: one 16x16 G or U tile per wave, K over D ----------
    // 16-wide body: group the four global b-loads + two 2ADDR a-loads ahead
    // of the four dependent WMMAs so they issue as one clause.
    {
      const float* wrow = W1 + (size_t)(ib + sub * 16 + mrow) * D_ + koff;
      if (ib + IK < I_)   // prefetch next superchunk's weight row (gfx1250)
        __builtin_prefetch(W1 + (size_t)(ib + IK + sub * 16 + mrow) * D_ + koff, 0, 1);
      const float* arow = xrowA + koff;
      v8f acc = (v8f){0,0,0,0,0,0,0,0};
      for (int k = 0; k < D_; k += 16) {
        const v2f b0 = *(const v2f*)(wrow + k);
        const v2f b1 = *(const v2f*)(wrow + k + 4);
        const v2f b2 = *(const v2f*)(wrow + k + 8);
        const v2f b3 = *(const v2f*)(wrow + k + 12);
        const v2f a0 = *(const v2f*)(arow + k);
        const v2f a1 = *(const v2f*)(arow + k + 4);
        const v2f a2 = *(const v2f*)(arow + k + 8);
        const v2f a3 = *(const v2f*)(arow + k + 12);
        acc = WMMA_F32X4(acc, a0, b0);
        acc = WMMA_F32X4(acc, a1, b1);
        acc = WMMA_F32X4(acc, a2, b2);
        acc = WMMA_F32X4(acc, a3, b3);
      }
      float* dst = GU + wave * 256;
#pragma unroll
      for (int v = 0; v < 8; ++v) dst[(v + half) * 16 + mrow] = acc[v];
    }
    __syncthreads();

    // ---------- h = silu(G) * U -> Hs (A-matrix layout for phase 2) --------
    {
#pragma unroll
      for (int j = 0; j < (BM * IK) / THREADS; ++j) {
        const int p = tid + j * THREADS;            // 0 .. 2047
        const int m = p >> 7;                       // token row
        const int i = p & 127;                      // i within superchunk
        const int sc = i >> 4, n = i & 15;
        const float g = GU[(sc * 2 + 0) * 256 + m * 16 + n];
        const float u = GU[(sc * 2 + 1) * 256 + m * 16 + n];
        Hs[m * HS_STRIDE + i] = (g / (1.0f + expf(-g))) * u;
      }
    }
#if USE_TDM
    if (wave == 0) __builtin_amdgcn_s_wait_tensorcnt(0);  // Wd tile landed
#endif
    __syncthreads();   // publish Hs (and TDM tile) to all waves

    // ---------- phase 2: OUT(16x128 per wave) += H(16x128) @ Wd^T ----------
    {
      const int dbase = wave * 128;
      const float* hrow = Hs + mrow * HS_STRIDE + koff;
      for (int k = 0; k < IK; k += 4) {
        const v2f a = *(const v2f*)(hrow + k);
        v2f b[8];
#pragma unroll
        for (int tn = 0; tn < 8; ++tn) {
#if USE_TDM
          const float* wdrow =
              WdLds + (size_t)(dbase + tn * 16 + mrow) * WD_STRIDE + koff;
          b[tn] = *(const v2f*)(wdrow + k);
#else
          const float* wdrow =
              Wd + (size_t)(dbase + tn * 16 + mrow) * I_ + ib + koff;
          b[tn] = *(const v2f*)(wdrow + k);
#endif
        }
#pragma unroll
        for (int tn = 0; tn < 8; ++tn) oacc[tn] = WMMA_F32X4(oacc[tn], a, b[tn]);
      }
    }
    __syncthreads();  // Hs/GU (and WdLds) reusable next superchunk
  }

  // ---------- scale by combine weight, accumulate into output -------------
#pragma unroll
  for (int tn = 0; tn < 8; ++tn) {
    const int col = wave * 128 + tn * 16 + mrow;
#pragma unroll
    for (int v = 0; v < 8; ++v) {
      const int m = v + half;
      const float val = oacc[tn][v] * cw_s[m];
      atomicAdd(out + (size_t)tok_s[m] * D_ + col, val);
    }
  }
}

// ---------------------------------------------------------------------------
extern "C" void kernel_launch(void* const* d_in, const int* in_sizes, int n_in,
                              void* d_out, int out_size, void* d_ws, size_t ws_size,
                              hipStream_t stream) {
  (void)in_sizes; (void)n_in; (void)out_size; (void)ws_size;

  const float* x  = (const float*)d_in[0];   // (B, L, D) fp32
  const float* gw = (const float*)d_in[1];   // (E, D)
  const float* eg = (const float*)d_in[2];   // (E, I, D)
  const float* eu = (const float*)d_in[3];   // (E, I, D)
  const float* ed = (const float*)d_in[4];   // (E, D, I)
  const float* sg = (const float*)d_in[5];   // (I, D)
  const float* su = (const float*)d_in[6];   // (I, D)
  const float* sd = (const float*)d_in[7];   // (D, I)
  float* out = (float*)d_out;                // T*D floats + 1 aux scalar

  char* ws = (char*)d_ws;
  int*   cnt  = (int*)ws;                                 // E counters
  float* ps   = (float*)(ws + 64);                        // E prob sums
  int*   idxl = (int*)(ws + 128);                         // E*T gather indices
  float* wlst = (float*)(ws + 128 + (size_t)E_ * T_ * 4); // E*T combine wts

  const size_t nOut = (size_t)T_ * D_ + 1;
  moe_zero_kernel<<<4096, 256, 0, stream>>>(out, nOut, cnt, ps);

  moe_router_kernel<<<T_ / 8, 256, 0, stream>>>(x, gw, cnt, ps, idxl, wlst);

  moe_aux_kernel<<<1, 32, 0, stream>>>(cnt, ps, out + (size_t)T_ * D_);

  const size_t smemBytes = (size_t)SMEM_FLOATS * sizeof(float);  // ~304 KB
  dim3 grid(T_ / BM, E_ + 1);  // tiles x (experts + shared)
  moe_mlp_kernel<<<grid, THREADS, smemBytes, stream>>>(
      x, eg, eu, ed, sg, su, sd, idxl, cnt, wlst, out);
}